// PhaseRefinement_3195455668481
// MI455X (gfx1250) — compile-verified
//
#include <hip/hip_runtime.h>
#include <math.h>

typedef __attribute__((ext_vector_type(2))) float v2f;
typedef __attribute__((ext_vector_type(4))) float v4f;
typedef __attribute__((ext_vector_type(8))) float v8f;

#define DD   4096
#define PP   16
#define TM   16         // rows per tile
#define NWAVES 8
#define KCHUNK (DD / NWAVES)   // 512
#define PI_F 3.14159265358979323846f

__launch_bounds__(256)
__global__ void phase_refine_fused(const float* __restrict__ x,
                                   const float* __restrict__ Wr,
                                   const float* __restrict__ br,
                                   const float* __restrict__ Wo,
                                   const float* __restrict__ bo,
                                   const float* __restrict__ gamma,
                                   const float* __restrict__ beta,
                                   float* __restrict__ out)
{
    __shared__ float redbuf[NWAVES][2][8][32];  // per-wave partial C tiles
    __shared__ float dotm[2][TM][PP];           // reduced dot products
    __shared__ float cosd[TM][PP];
    __shared__ float xsum[TM];
    __shared__ float xsq[TM];
    __shared__ float rowA[TM];
    __shared__ float rowB[TM];

    const int t = threadIdx.x;
    const int w = __builtin_amdgcn_readfirstlane(t >> 5);  // wave id, scalar
    const int l = t & 31;        // lane
    const int m = l & 15;        // row within tile (A) / plane index (B)
    const int h = l >> 4;        // half-wave
    const long long r0 = (long long)blockIdx.x * TM;

    if (t < TM) { xsum[t] = 0.0f; xsq[t] = 0.0f; }
    __syncthreads();

    // ---- Phase 1: dual WMMA GEMM over this wave's K-chunk -----------------
    v8f accR = {0.f,0.f,0.f,0.f,0.f,0.f,0.f,0.f};
    v8f accO = {0.f,0.f,0.f,0.f,0.f,0.f,0.f,0.f};
    float s = 0.0f, sq = 0.0f;

    const float* xrow  = x  + (r0 + m) * DD;          // A: row m of tile
    const float* wrrow = Wr + (long long)m * DD;      // B: plane m (N = lane%16)
    const float* worow = Wo + (long long)m * DD;
    const int k0 = w * KCHUNK;                        // scalar bound -> uniform loop

    #pragma unroll 8
    for (int k = k0; k < k0 + KCHUNK; k += 4) {
        // A 16x4 f32: lane holds K = {k+2h, k+2h+1} -> one b64 load
        v2f a  = *(const v2f*)(xrow  + k + 2*h);
        v2f bR = *(const v2f*)(wrrow + k + 2*h);
        v2f bO = *(const v2f*)(worow + k + 2*h);
        accR = __builtin_amdgcn_wmma_f32_16x16x4_f32(
                   false, a, false, bR, (short)0, accR, false, false);
        accO = __builtin_amdgcn_wmma_f32_16x16x4_f32(
                   false, a, false, bO, (short)0, accO, false, false);
        s  += a.x + a.y;              // row moment accumulation (each elem once)
        sq += a.x*a.x + a.y*a.y;
    }
    atomicAdd(&xsum[m], s);           // ds_add_f32
    atomicAdd(&xsq[m],  sq);
    #pragma unroll
    for (int v = 0; v < 8; ++v) {
        redbuf[w][0][v][l] = accR[v];
        redbuf[w][1][v][l] = accO[v];
    }
    __syncthreads();

    // ---- Cross-wave C reduction: 512 (matrix,vgpr,lane) combos ------------
    #pragma unroll
    for (int c = 0; c < 2; ++c) {
        int combo = t + c * 256;
        int mat = combo >> 8;
        int v   = (combo >> 5) & 7;
        int ll  = combo & 31;
        float acc = 0.0f;
        #pragma unroll
        for (int ww = 0; ww < NWAVES; ++ww) acc += redbuf[ww][mat][v][ll];
        int M = v + 8 * (ll >> 4);    // C/D layout: VGPR v, half -> M
        int N = ll & 15;
        dotm[mat][M][N] = acc;
    }
    __syncthreads();

    // ---- Phases -> cos(r - o), one (M,N) per thread -----------------------
    {
        int M = t >> 4, N = t & 15;
        float pr = tanhf(dotm[0][M][N] + br[N]) * PI_F;
        float po = tanhf(dotm[1][M][N] + bo[N]) * PI_F;
        cosd[M][N] = cosf(pr - po);
    }
    __syncthreads();

    // ---- Per-row scalars: gain, fused LayerNorm coefficients --------------
    if (t < TM) {
        float S = 0.0f;
        #pragma unroll
        for (int n = 0; n < PP; ++n) S += cosd[t][n];
        float reso = S * (1.0f / PP);
        float z    = reso + 0.5f;
        float gain = log1pf(__expf(z));        // softplus, z in [-0.5, 1.5]
        float cc   = S * gain * (1.0f / PP);   // output = cc * x
        float mx   = xsum[t] * (1.0f / DD);
        float ex2  = xsq[t]  * (1.0f / DD);
        float var  = ex2 - mx * mx;
        float inv  = rsqrtf(cc * cc * var + 1e-5f);
        rowA[t] =  cc * inv;
        rowB[t] = -cc * mx * inv;
    }
    __syncthreads();

    // ---- Phase 2: elementwise epilogue, tile is hot in cache --------------
    // x is last-use here (NT load); out is write-once (NT store) so it does
    // not evict x/W working set from L2.
    const v4f* x4 = (const v4f*)(x   + r0 * DD);
    v4f*       o4 = (v4f*)      (out + r0 * DD);
    const v4f* g4 = (const v4f*)gamma;
    const v4f* b4 = (const v4f*)beta;
    #pragma unroll 4
    for (int i = 0; i < (TM * DD / 4) / 256; ++i) {   // 64 iterations
        int f4   = t + i * 256;        // float4 index within tile (coalesced)
        int row  = f4 >> 10;           // 1024 float4 per row
        int col4 = f4 & 1023;
        float A  = rowA[row];
        float Bv = rowB[row];
        v4f xv = __builtin_nontemporal_load(&x4[f4]);
        v4f gv = g4[col4];
        v4f bv = b4[col4];
        v4f ov;
        ov.x = xv.x * (1.0f + A * gv.x) + Bv * gv.x + bv.x;
        ov.y = xv.y * (1.0f + A * gv.y) + Bv * gv.y + bv.y;
        ov.z = xv.z * (1.0f + A * gv.z) + Bv * gv.z + bv.z;
        ov.w = xv.w * (1.0f + A * gv.w) + Bv * gv.w + bv.w;
        __builtin_nontemporal_store(ov, &o4[f4]);
    }
}

extern "C" void kernel_launch(void* const* d_in, const int* in_sizes, int n_in,
                              void* d_out, int out_size, void* d_ws, size_t ws_size,
                              hipStream_t stream) {
    const float* x     = (const float*)d_in[0];
    const float* Wr    = (const float*)d_in[1];
    const float* brf   = (const float*)d_in[2];
    const float* Wo    = (const float*)d_in[3];
    const float* bout  = (const float*)d_in[4];
    const float* gamma = (const float*)d_in[5];
    const float* beta  = (const float*)d_in[6];
    float* out = (float*)d_out;

    const int Brows = in_sizes[0] / DD;       // 32768
    const int grid  = Brows / TM;             // 2048 workgroups
    hipLaunchKernelGGL(phase_refine_fused, dim3(grid), dim3(256), 0, stream,
                       x, Wr, brf, Wo, bout, gamma, beta, out);
}